// ACSConv_29480655520371
// MI455X (gfx1250) — compile-verified
//
#include <hip/hip_runtime.h>

// CDNA5 / gfx1250, wave32. Uses V_WMMA_F32_16X16X4_F32 for both the big
// streaming matmul (memory-bound: 14 x 1GB of Ls @ 23.3 TB/s) and the small
// projection GEMMs. f32 kept end-to-end for the 15-step Chebyshev recurrence.

typedef __attribute__((ext_vector_type(2))) float v2f;
typedef __attribute__((ext_vector_type(8))) float v8f;

#define NA    16384   // N_ANGLES * N
#define NROWS 2048    // N
#define NCOL  16      // Cin
#define COUT  32
#define KCH   15

// ---------------------------------------------------------------------------
// Tout[16384,16] = alpha-less Chebyshev step:
//   Tprev == nullptr :  Tout = Ls @ Tin
//   else             :  Tout = 2*(Ls @ Tin) - Tprev
// Grid: 1024 blocks (one 16-row band of Ls), 256 threads = 8 waves.
// Each wave covers a 2048-wide K slice with 512 WMMA f32 16x16x4 ops,
// partials reduced through LDS.
// ---------------------------------------------------------------------------
__global__ __launch_bounds__(256) void cheb_step(
    float* __restrict__ Tout,
    const float* __restrict__ Ls,
    const float* __restrict__ Tin,
    const float* __restrict__ Tprev)
{
    __shared__ float red[8 * 256];      // 8 waves x 16x16 f32 partial tiles

    const int lane  = threadIdx.x & 31;
    const int wave  = threadIdx.x >> 5;
    const int mtile = blockIdx.x;               // 0..1023
    const int m     = lane & 15;                // A-matrix row within tile
    const int kgrp  = lane >> 4;                // 0: K{0,1}, 1: K{2,3}
    const int row   = mtile * 16 + m;
    const int kstart = wave * 2048;

    // A: 16x4 f32 layout -> lane loads float2 (its K pair) from its Ls row.
    const float* aptr = Ls + (size_t)row * NA + (size_t)(kstart + 2 * kgrp);
    // B: 4x16 f32 layout -> lane (N = lane&15) loads rows k0 and k0+1 of Tin.
    const float* bptr = Tin + (size_t)(kstart + 2 * kgrp) * NCOL + m;

    v8f c = {};
    #pragma unroll 8
    for (int kk = 0; kk < 2048; kk += 4) {
        if ((kk & 31) == 0)
            __builtin_prefetch(aptr + 256, 0, 0);   // 1KB ahead on the Ls stream
        v2f a = *(const v2f*)aptr;                  // Ls[row, k0], Ls[row, k0+1]
        v2f b;
        b.x = bptr[0];                              // Tin[k0,   n]
        b.y = bptr[NCOL];                           // Tin[k0+1, n]
        c = __builtin_amdgcn_wmma_f32_16x16x4_f32(
                false, a, false, b, (short)0, c, false, false);
        aptr += 4;
        bptr += 4 * NCOL;
    }

    // Reduce the 8 per-wave partial 16x16 tiles.
    #pragma unroll
    for (int i = 0; i < 8; ++i)
        red[wave * 256 + i * 32 + lane] = c[i];
    __syncthreads();

    const int t = threadIdx.x;          // one output element per thread
    float sum = 0.0f;
    #pragma unroll
    for (int w = 0; w < 8; ++w)
        sum += red[w * 256 + t];

    // Decode C/D layout: VGPR i, lanes 0-15 -> M=i, lanes 16-31 -> M=i+8.
    const int i  = t >> 5;
    const int l  = t & 31;
    const int mm = i + ((l & 16) ? 8 : 0);
    const int nn = l & 15;
    const size_t oidx = (size_t)(mtile * 16 + mm) * NCOL + nn;

    Tout[oidx] = Tprev ? (2.0f * sum - Tprev[oidx]) : sum;
}

// ---------------------------------------------------------------------------
// out[2048,32] += permute(Tx)[2048,128] @ Wk[128,32]
// where permute(Tx)[r, a*16+c] = Tx[a*2048 + r, c].
// Grid: 256 blocks of 1 wave; block = (mtile, ntile) 16x16 output tile.
// ---------------------------------------------------------------------------
__global__ __launch_bounds__(32) void proj_acc(
    float* __restrict__ out,
    const float* __restrict__ Tx,
    const float* __restrict__ Wk)
{
    const int lane  = threadIdx.x & 31;
    const int mtile = blockIdx.x >> 1;   // 0..127
    const int ntile = blockIdx.x & 1;    // 0..1
    const int m     = lane & 15;
    const int kgrp  = lane >> 4;
    const int r     = mtile * 16 + m;

    v8f c = {};
    #pragma unroll
    for (int kk = 0; kk < 128; kk += 4) {
        const int j0 = kk + 2 * kgrp;            // even -> pair stays in one angle
        const int a0 = j0 >> 4;                  // angle index
        const int c0 = j0 & 15;                  // channel index (even)
        v2f a = *(const v2f*)(Tx + ((size_t)(a0 * NROWS + r)) * NCOL + c0);
        v2f b;
        b.x = Wk[(j0    ) * COUT + ntile * 16 + m];
        b.y = Wk[(j0 + 1) * COUT + ntile * 16 + m];
        c = __builtin_amdgcn_wmma_f32_16x16x4_f32(
                false, a, false, b, (short)0, c, false, false);
    }

    #pragma unroll
    for (int i = 0; i < 8; ++i) {
        const int mm = i + ((lane & 16) ? 8 : 0);
        const int nn = lane & 15;
        const size_t oidx = (size_t)(mtile * 16 + mm) * COUT + ntile * 16 + nn;
        out[oidx] += c[i];                       // sequential launches: no races
    }
}

// out[r, c] = bias[c]  (fold bias into the accumulator init)
__global__ void init_out(float* __restrict__ out, const float* __restrict__ bias)
{
    const int i = blockIdx.x * 256 + threadIdx.x;
    if (i < NROWS * COUT) out[i] = bias[i & (COUT - 1)];
}

// T0 = tile(x, (8,1)):  T0 flat[i] = x flat[i mod (2048*16)]
__global__ void init_tx0(float* __restrict__ T0, const float* __restrict__ x)
{
    const int i = blockIdx.x * 256 + threadIdx.x;
    if (i < NA * NCOL) T0[i] = x[i & (NROWS * NCOL - 1)];
}

extern "C" void kernel_launch(void* const* d_in, const int* in_sizes, int n_in,
                              void* d_out, int out_size, void* d_ws, size_t ws_size,
                              hipStream_t stream)
{
    (void)in_sizes; (void)n_in; (void)out_size; (void)ws_size;

    const float* x    = (const float*)d_in[0];   // [2048, 16]
    const float* Ls   = (const float*)d_in[1];   // [16384, 16384]
    const float* W    = (const float*)d_in[2];   // [15, 128, 32]
    const float* bias = (const float*)d_in[3];   // [32]
    float* out = (float*)d_out;                  // [2048, 32]

    // Three rotating Chebyshev operand buffers (1 MB each) in workspace.
    float* T0 = (float*)d_ws;
    float* T1 = T0 + (size_t)NA * NCOL;
    float* T2 = T1 + (size_t)NA * NCOL;

    const int WSTRIDE = (NCOL * 8) * COUT;       // 128*32 floats per weight slab

    init_out<<<(NROWS * COUT + 255) / 256, 256, 0, stream>>>(out, bias);
    init_tx0<<<(NA * NCOL + 255) / 256, 256, 0, stream>>>(T0, x);

    // k = 0
    proj_acc<<<256, 32, 0, stream>>>(out, T0, W);
    // k = 1: T1 = Ls @ T0
    cheb_step<<<1024, 256, 0, stream>>>(T1, Ls, T0, nullptr);
    proj_acc<<<256, 32, 0, stream>>>(out, T1, W + WSTRIDE);
    // k = 2..14: T2 = 2*(Ls @ T1) - T0
    for (int k = 2; k < KCH; ++k) {
        cheb_step<<<1024, 256, 0, stream>>>(T2, Ls, T1, T0);
        proj_acc<<<256, 32, 0, stream>>>(out, T2, W + (size_t)k * WSTRIDE);
        float* tmp = T0; T0 = T1; T1 = T2; T2 = tmp;
    }
}